// Detect_30846455120277
// MI455X (gfx1250) — compile-verified
//
#include <hip/hip_runtime.h>
#include <cstdint>
#include <cfloat>

// ---------------------------------------------------------------------------
// SSD Detect post-process for MI455X (gfx1250).
//  B=16, A=16384 anchors, C=81 classes, TOP_K=64.
//  Kernel 1: decode (loc, anchors) -> corner boxes in d_ws (B*A float4 = 4MB)
//  Kernel 2: one workgroup per (batch, class):
//     - TDM tensor_load_to_lds: strided conf column (stride 324B) -> LDS
//       (one wave-level DMA instruction, TENSORcnt), fallback = per-lane
//       global_load_async_to_lds_b32 gather
//     - decoded boxes stream in via global_load_async_to_lds_b128 (ASYNCcnt)
//     - 64 rounds of greedy NMS entirely out of LDS/registers:
//       wave32 __shfl_xor argmax + cross-wave LDS reduce, IoU suppression.
//  Scores stay in LOGIT domain (sigmoid monotone): threshold = ln(0.3/0.7);
//  sigmoid evaluated only for emitted rows.
// ---------------------------------------------------------------------------

#define A_NUM 16384
#define B_NUM 16
#define C_NUM 81
#define TOPK  64
#define TPB   1024                 // 32 wave32s per workgroup
#define PT    (A_NUM / TPB)        // 16 anchors per thread
#define LOGIT_THRESH (-0.8472978603872036f)   // ln(0.3/0.7)
#define NMS_T 0.5f
// LDS: boxes float4[A] (256KB) + score staging float[A] (64KB) = 320KB (CDNA5 max)
#define SMEM_BYTES (A_NUM * 16 + A_NUM * 4)

#if defined(__has_builtin)
#if __has_builtin(__builtin_amdgcn_tensor_load_to_lds) && \
    __has_builtin(__builtin_amdgcn_s_wait_tensorcnt)
#define HAVE_TDM 1
#endif
#endif

#ifdef HAVE_TDM
typedef unsigned int u32x4 __attribute__((ext_vector_type(4)));
typedef int          i32x8 __attribute__((ext_vector_type(8)));
typedef int          i32x4 __attribute__((ext_vector_type(4)));
#endif

__global__ __launch_bounds__(256) void decode_kernel(
    const float* __restrict__ loc, const float* __restrict__ anchors,
    float4* __restrict__ boxes)
{
    int t = blockIdx.x * blockDim.x + threadIdx.x;     // t in [0, B*A)
    int a = t & (A_NUM - 1);
    float4 l  = ((const float4*)loc)[t];
    float4 an = ((const float4*)anchors)[a];
    float cx = fmaf(l.x * 0.1f, an.z, an.x);
    float cy = fmaf(l.y * 0.1f, an.w, an.y);
    float w  = an.z * expf(l.z * 0.2f);
    float h  = an.w * expf(l.w * 0.2f);
    float4 o;
    o.x = cx - 0.5f * w;
    o.y = cy - 0.5f * h;
    o.z = o.x + w;
    o.w = o.y + h;
    boxes[t] = o;
}

__global__ __launch_bounds__(TPB) void nms_kernel(
    const float* __restrict__ conf,        // [B, A, C] logits
    const float4* __restrict__ boxes_g,    // [B, A] decoded corners (workspace)
    float* __restrict__ out)               // [B, C, TOPK, 5]
{
    extern __shared__ char smem[];
    float4* bx    = (float4*)smem;                                  // [A] boxes
    float*  stage = (float*)(smem + (size_t)A_NUM * sizeof(float4));// [A] scores
    // after scores move to registers, the first words of `stage` are reused:
    float* red_v = stage;               // [32]  per-wave argmax value
    int*   red_i = (int*)(stage + 32);  // [32]  per-wave argmax index
    float* bc    = stage + 64;          // [6]   broadcast: score, box, idx

    const int c   = blockIdx.x;
    const int b   = blockIdx.y;
    const int tid = threadIdx.x;
    const int wave = tid >> 5, lane = tid & 31;

    // ---- stage scores conf[b, a, c] (stride C floats) -> stage[a] ----------
#ifdef HAVE_TDM
    if (wave == 0) {
        // Tensor DMA descriptor (ISA cdna5 ch.8): 2D tensor, tile = 1 x 16384,
        // dim0 stride = C elements, data_size = 4B, contiguous into LDS.
        unsigned long long ga =
            (unsigned long long)(uintptr_t)(conf + (size_t)b * A_NUM * C_NUM + c);
        unsigned lds_addr = (unsigned)(uintptr_t)stage;
        u32x4 g0;
        g0.x = 1u;                                   // count=1 valid, no gather
        g0.y = lds_addr;                             // lds_addr [63:32]
        g0.z = (unsigned)(ga & 0xffffffffu);         // global_addr [95:64]
        g0.w = (unsigned)((ga >> 32) & 0x1ffffffu)   // global_addr [120:96]
             | (2u << 30);                           // type=2 ("image")
        i32x8 g1;
        g1.s0 = (int)(2u << 16);          // wg_mask=0, data_size=2 (4 bytes)
        g1.s1 = (int)((unsigned)C_NUM << 16);   // tensor_dim0[15:0] = 81
        g1.s2 = (int)((unsigned)A_NUM << 16);   // tensor_dim1[15:0] = 16384
        g1.s3 = (int)(1u << 16);          // tile_dim0 = 1 element
        g1.s4 = (int)(unsigned)A_NUM;     // tile_dim1 = 16384, tile_dim2 = 0
        g1.s5 = (int)(unsigned)C_NUM;     // tensor_dim0_stride = 81 elements
        g1.s6 = 0;                        // stride hi / tensor_dim1_stride lo
        g1.s7 = 0;                        // tensor_dim1_stride hi
        i32x4 gz4 = {0, 0, 0, 0};             // groups 2/3 unused (2D tensor)
        i32x8 gz8 = {0, 0, 0, 0, 0, 0, 0, 0}; // trailing group unused
        __builtin_amdgcn_tensor_load_to_lds(g0, g1, gz4, gz4, gz8, 0);
    }
#else
    {
        unsigned long long gbase =
            (unsigned long long)(uintptr_t)(conf + (size_t)b * A_NUM * C_NUM + c);
        for (int i = 0; i < PT; ++i) {
            int a = tid + i * TPB;
            unsigned lds = (unsigned)(uintptr_t)(&stage[a]);
            unsigned long long g = gbase + (unsigned long long)a * (C_NUM * 4);
            asm volatile("global_load_async_to_lds_b32 %0, %1, off"
                         :: "v"(lds), "v"(g) : "memory");
        }
    }
#endif
    // ---- async stage: boxes_g[b, a] -> bx[a] (coalesced b128) --------------
    {
        unsigned long long gbase =
            (unsigned long long)(uintptr_t)(boxes_g + (size_t)b * A_NUM);
        for (int i = 0; i < PT; ++i) {
            int a = tid + i * TPB;
            unsigned lds = (unsigned)(uintptr_t)(&bx[a]);
            unsigned long long g = gbase + (unsigned long long)a * 16;
            asm volatile("global_load_async_to_lds_b128 %0, %1, off"
                         :: "v"(lds), "v"(g) : "memory");
        }
    }
    asm volatile("s_wait_asynccnt 0x0" ::: "memory");
#ifdef HAVE_TDM
    if (wave == 0) __builtin_amdgcn_s_wait_tensorcnt(0);
#endif
    __syncthreads();

    // ---- scores -> registers, confidence threshold in logit domain ---------
    float sc[PT];
    for (int i = 0; i < PT; ++i) {
        float v = stage[tid + i * TPB];
        sc[i] = (v > LOGIT_THRESH) ? v : -FLT_MAX;
    }
    __syncthreads();   // stage[] now reusable as reduction scratch

    float* outp = out + ((size_t)(b * C_NUM + c) * TOPK) * 5;

    int k = 0;
    for (; k < TOPK; ++k) {
        // thread-local argmax (first-max tie-break, like jnp.argmax)
        float bv = -FLT_MAX;
        int   bi = 0x7fffffff;
        for (int i = 0; i < PT; ++i) {
            int a = tid + i * TPB;
            if (sc[i] > bv || (sc[i] == bv && a < bi)) { bv = sc[i]; bi = a; }
        }
        // wave32 reduction
        for (int off = 16; off >= 1; off >>= 1) {
            float ov = __shfl_xor(bv, off, 32);
            int   oi = __shfl_xor(bi, off, 32);
            if (ov > bv || (ov == bv && oi < bi)) { bv = ov; bi = oi; }
        }
        if (lane == 0) { red_v[wave] = bv; red_i[wave] = bi; }
        __syncthreads();
        if (wave == 0) {
            bv = red_v[lane];
            bi = red_i[lane];
            for (int off = 16; off >= 1; off >>= 1) {
                float ov = __shfl_xor(bv, off, 32);
                int   oi = __shfl_xor(bi, off, 32);
                if (ov > bv || (ov == bv && oi < bi)) { bv = ov; bi = oi; }
            }
            if (lane == 0) {
                bc[0] = bv;
                float4 cb = bx[bi];
                bc[1] = cb.x; bc[2] = cb.y; bc[3] = cb.z; bc[4] = cb.w;
                bc[5] = __int_as_float(bi);
            }
        }
        __syncthreads();

        float champ = bc[0];
        if (!(champ > LOGIT_THRESH)) break;   // uniform decision: all done
        float x1 = bc[1], y1 = bc[2], x2 = bc[3], y2 = bc[4];
        int   ci = __float_as_int(bc[5]);

        if (tid == 0) {
            float s = 1.0f / (1.0f + expf(-champ));   // sigmoid only for winners
            outp[k * 5 + 0] = s;
            outp[k * 5 + 1] = x1;
            outp[k * 5 + 2] = y1;
            outp[k * 5 + 3] = x2;
            outp[k * 5 + 4] = y2;
        }

        // IoU suppression against LDS-resident boxes
        float a1 = (x2 - x1) * (y2 - y1);
        for (int i = 0; i < PT; ++i) {
            int a = tid + i * TPB;
            float4 q = bx[a];
            float tlx = fmaxf(x1, q.x), tly = fmaxf(y1, q.y);
            float brx = fminf(x2, q.z), bry = fminf(y2, q.w);
            float w = fmaxf(brx - tlx, 0.0f), h = fmaxf(bry - tly, 0.0f);
            float inter = w * h;
            float a2 = (q.z - q.x) * (q.w - q.y);
            float iou = inter / (a1 + a2 - inter);
            if (iou > NMS_T || a == ci) sc[i] = -FLT_MAX;
        }
        // no barrier needed here: red/bc writes next iter are fenced by the
        // __syncthreads() after the red_v/red_i stores.
    }

    // zero-fill remaining rows (output is poisoned by the harness)
    for (int r = k * 5 + tid; r < TOPK * 5; r += TPB) outp[r] = 0.0f;
}

extern "C" void kernel_launch(void* const* d_in, const int* in_sizes, int n_in,
                              void* d_out, int out_size, void* d_ws, size_t ws_size,
                              hipStream_t stream)
{
    (void)in_sizes; (void)n_in; (void)out_size; (void)ws_size;
    const float* loc     = (const float*)d_in[0];   // [B, A, 4]
    const float* conf    = (const float*)d_in[1];   // [B, A, C]
    const float* anchors = (const float*)d_in[2];   // [A, 4]
    float* out   = (float*)d_out;                   // [B, C, TOPK, 5]
    float4* boxes = (float4*)d_ws;                  // B*A float4 = 4 MiB scratch

    (void)hipFuncSetAttribute(reinterpret_cast<const void*>(nms_kernel),
                              hipFuncAttributeMaxDynamicSharedMemorySize,
                              SMEM_BYTES);

    decode_kernel<<<(B_NUM * A_NUM) / 256, 256, 0, stream>>>(loc, anchors, boxes);

    dim3 grid(C_NUM, B_NUM);   // 81 x 16 = 1296 independent NMS lanes
    nms_kernel<<<grid, TPB, SMEM_BYTES, stream>>>(conf, boxes, out);
}